// BlockDiagonalLinear_50113678409894
// MI455X (gfx1250) — compile-verified
//
#include <hip/hip_runtime.h>

typedef float v2f __attribute__((ext_vector_type(2)));
typedef float v8f __attribute__((ext_vector_type(8)));

#define WIDTH   4096
#define NB      16
#define BW      256      // block width (per-block N and K)
#define KC      64       // K chunk staged per iteration
#define ASTRIDE 68       // 64 + 4 pad: b64 A reads hit disjoint bank quads
#define BSTRIDE 136      // 128 + 8 pad: lane halves hit disjoint bank sets
#define A_BUF   (64 * ASTRIDE)   // floats per A buffer (4352)
#define B_BUF   (KC * BSTRIDE)   // floats per B buffer (8704)

__global__ __launch_bounds__(256)
void BlockDiagonalLinear_kernel(const float* __restrict__ x,
                                const float* __restrict__ w,
                                const float* __restrict__ b,
                                float* __restrict__ out)
{
    extern __shared__ float lds[];
    float* As = lds;               // 2 * A_BUF
    float* Bs = lds + 2 * A_BUF;   // 2 * B_BUF

    const int tid   = threadIdx.x;
    const int wave  = tid >> 5;
    const int lane  = tid & 31;
    const int lhalf = lane >> 4;     // 0: lanes 0-15, 1: lanes 16-31
    const int l16   = lane & 15;
    const int mwave = wave & 3;      // 4 M sub-tiles
    const int nwave = wave >> 2;     // 2 N sub-tiles
    const int h     = blockIdx.z;    // diagonal block
    const int wgM   = blockIdx.x;    // 64-row tile index
    const int wgN   = blockIdx.y;    // 128-col half index

    const int m0   = wgM * 64 + mwave * 16;  // wave's first output row
    const int colg = wgN * 128;              // WG col offset inside block h
    const int ncl  = nwave * 64;             // wave col offset inside staged 128

    const float* xg = x + (size_t)wgM * 64 * WIDTH + (size_t)h * BW;
    const float* wg = w + (size_t)h * BW * BW + colg;

    float4 areg[4];
    float4 breg[8];

    auto loadA = [&](int kc) {
        #pragma unroll
        for (int i = 0; i < 4; ++i) {
            int idx = tid + i * 256;
            int r = idx >> 4, c4 = idx & 15;          // 16 float4 per 64-f row
            areg[i] = *(const float4*)(xg + (size_t)r * WIDTH + kc * KC + c4 * 4);
        }
    };
    auto loadB = [&](int kc) {
        #pragma unroll
        for (int i = 0; i < 8; ++i) {
            int idx = tid + i * 256;
            int r = idx >> 5, c4 = idx & 31;          // 32 float4 per 128-f row
            breg[i] = *(const float4*)(wg + (size_t)(kc * KC + r) * BW + c4 * 4);
        }
    };
    auto stage = [&](int buf) {
        float* Ab = As + buf * A_BUF;
        float* Bb = Bs + buf * B_BUF;
        #pragma unroll
        for (int i = 0; i < 4; ++i) {
            int idx = tid + i * 256;
            int r = idx >> 4, c4 = idx & 15;
            *(float4*)(Ab + r * ASTRIDE + c4 * 4) = areg[i];
        }
        #pragma unroll
        for (int i = 0; i < 8; ++i) {
            int idx = tid + i * 256;
            int r = idx >> 5, c4 = idx & 31;
            *(float4*)(Bb + r * BSTRIDE + c4 * 4) = breg[i];
        }
    };

    v8f acc[4] = {};

    const int arow = mwave * 16 + l16;   // this lane's A row inside the 64-row tile

    loadA(0);
    loadB(0);
    stage(0);
    __syncthreads();

    for (int kc = 0; kc < BW / KC; ++kc) {
        const int cur = kc & 1;
        if (kc + 1 < BW / KC) { loadA(kc + 1); loadB(kc + 1); }  // prefetch in flight

        const float* Ab = As + cur * A_BUF;
        const float* Bb = Bs + cur * B_BUF;

        #pragma unroll
        for (int kk = 0; kk < KC; kk += 4) {
            // A fragment (16x4 f32): v0 = K {kk | kk+2}, v1 = K {kk+1 | kk+3}
            v2f a = *(const v2f*)(Ab + arow * ASTRIDE + kk + 2 * lhalf);
            #pragma unroll
            for (int t = 0; t < 4; ++t) {
                int nc = ncl + t * 16 + l16;
                // B fragment (4x16 f32): rows kk..kk+3 split across lane halves
                v2f bf;
                bf.x = Bb[(kk + 0 + 2 * lhalf) * BSTRIDE + nc];
                bf.y = Bb[(kk + 1 + 2 * lhalf) * BSTRIDE + nc];
                acc[t] = __builtin_amdgcn_wmma_f32_16x16x4_f32(
                    /*neg_a=*/false, a, /*neg_b=*/false, bf,
                    /*c_mod=*/(short)0, acc[t],
                    /*reuse_a=*/false, /*reuse_b=*/false);
            }
        }

        if (kc + 1 < BW / KC) stage(1 - cur);
        __syncthreads();
    }

    // Epilogue: bias + store. C/D layout: VGPR r holds rows (r, r+8) across halves.
    #pragma unroll
    for (int t = 0; t < 4; ++t) {
        const int gc = h * BW + colg + ncl + t * 16 + l16;
        const float bv = b[h * BW + colg + ncl + t * 16 + l16];
        #pragma unroll
        for (int r = 0; r < 8; ++r) {
            out[(size_t)(m0 + r + 8 * lhalf) * WIDTH + gc] = acc[t][r] + bv;
        }
    }
}

extern "C" void kernel_launch(void* const* d_in, const int* in_sizes, int n_in,
                              void* d_out, int out_size, void* d_ws, size_t ws_size,
                              hipStream_t stream) {
    const float* x = (const float*)d_in[0];   // (4, 4096, 4096)
    const float* w = (const float*)d_in[1];   // (16, 256, 256)
    const float* b = (const float*)d_in[2];   // (16, 256)
    float* out = (float*)d_out;               // (4, 4096, 4096)

    const int rows = in_sizes[0] / WIDTH;     // 16384
    dim3 grid(rows / 64, 2, NB);
    dim3 block(256);
    size_t shmem = (size_t)(2 * A_BUF + 2 * B_BUF) * sizeof(float);  // ~102 KB

    BlockDiagonalLinear_kernel<<<grid, block, shmem, stream>>>(x, w, b, out);
}